// get_loss_84610855731466
// MI455X (gfx1250) — compile-verified
//
#include <hip/hip_runtime.h>
#include <hip/hip_bf16.h>
#include <stdint.h>

#define INSNUM 1024
#define NPAIRS (INSNUM * (INSNUM - 1))   // 1,047,552
#define NRELS  200000
#define NCLASS 27
#define THREADS 256
#define ROWS_PER_BLOCK 256
#define NBLOCKS (NPAIRS / ROWS_PER_BLOCK)        // 4092 (exact)
#define TILE_FLOATS (ROWS_PER_BLOCK * NCLASS)    // 6912
#define TILE_VEC4 (TILE_FLOATS / 4)              // 1728

typedef float v2f __attribute__((ext_vector_type(2)));
typedef float v8f __attribute__((ext_vector_type(8)));
typedef int   v4i __attribute__((ext_vector_type(4)));

typedef __attribute__((address_space(1))) v4i* g_v4i_ptr;  // global (AS1)
typedef __attribute__((address_space(3))) v4i* l_v4i_ptr;  // LDS (AS3)

#if __has_builtin(__builtin_amdgcn_global_load_async_to_lds_b128)
#define HAS_ASYNC_LDS 1
#endif
#if __has_builtin(__builtin_amdgcn_wmma_f32_16x16x4_f32)
#define HAS_WMMA_F32X4 1
#endif

__device__ __forceinline__ void wait_async_zero() {
#if __has_builtin(__builtin_amdgcn_s_wait_asynccnt)
    __builtin_amdgcn_s_wait_asynccnt(0);
#else
    asm volatile("s_wait_asynccnt 0" ::: "memory");
#endif
}

// ---------------------------------------------------------------------------
// Kernel 1: scatter relations into per-row 27-bit class masks (atomicOr is
// order-independent -> deterministic, and OR == repeated .set(1.0)).
// ---------------------------------------------------------------------------
__global__ void build_mask_kernel(const int* __restrict__ rel,
                                  unsigned* __restrict__ bits) {
    int r = blockIdx.x * blockDim.x + threadIdx.x;
    if (r >= NRELS) return;
    int i = rel[3 * r + 0];
    int j = rel[3 * r + 1];
    int c = rel[3 * r + 2];
    if (i == j) return;
    int flat = (i < j) ? (i * (INSNUM - 1) + j - 1) : (i * (INSNUM - 1) + j);
    atomicOr(&bits[flat], 1u << c);
}

// ---------------------------------------------------------------------------
// Kernel 2: stream 256 rows/block into LDS (async B128 path), per-thread
// softmax + masked focal loss, WMMA-based block reduction.
// ---------------------------------------------------------------------------
__global__ __launch_bounds__(THREADS) void row_loss_kernel(
        const float* __restrict__ pred,
        const float* __restrict__ alpha,
        const unsigned* __restrict__ bits,
        float* __restrict__ partial) {
    __shared__ float tile[TILE_FLOATS];
    __shared__ float s_alpha[NCLASS];
    __shared__ float wave_sums[THREADS / 32];

    const int tid = threadIdx.x;
    const int row0 = blockIdx.x * ROWS_PER_BLOCK;
    const float* gsrc = pred + (size_t)row0 * NCLASS;

    if (tid < NCLASS) s_alpha[tid] = alpha[tid];

#ifdef HAS_ASYNC_LDS
    for (int idx = tid; idx < TILE_VEC4; idx += THREADS) {
        __builtin_amdgcn_global_load_async_to_lds_b128(
            (g_v4i_ptr)(gsrc + 4 * idx),
            (l_v4i_ptr)(&tile[4 * idx]),
            0, 0);
    }
    wait_async_zero();
#else
    for (int idx = tid; idx < TILE_VEC4; idx += THREADS) {
        float4 v = ((const float4*)gsrc)[idx];
        ((float4*)tile)[idx] = v;
    }
#endif
    __syncthreads();

    // One row per thread; stride-27 dwords is bank-conflict-free (gcd(27,64)=1).
    const float* x = &tile[tid * NCLASS];
    unsigned m = bits[row0 + tid];
    if (m == 0u) m = 1u;  // background class

    float mx = x[0];
#pragma unroll
    for (int c = 1; c < NCLASS; ++c) mx = fmaxf(mx, x[c]);

    float e[NCLASS];
    float denom = 0.0f;
#pragma unroll
    for (int c = 0; c < NCLASS; ++c) {
        e[c] = __expf(x[c] - mx);
        denom += e[c];
    }

    float psum = 0.0f, asum = 0.0f;
#pragma unroll
    for (int c = 0; c < NCLASS; ++c) {
        if (m & (1u << c)) {
            psum += e[c];
            asum += s_alpha[c];
        }
    }

    float probs = psum / denom;                      // in (0, 1]
    float lp = __logf(fmaxf(probs, 1e-12f));
    float om = 1.0f - probs;
    float loss = -(asum * om * om * lp);             // GAMMA == 2

    // ---- wave reduction of 32 lane losses via WMMA f32 16x16x4 ----
    float s;
#ifdef HAS_WMMA_F32X4
    {
        // A(16x4): lane L<16 -> A[L][0]=v ; lane L>=16 -> A[L-16][2]=v.
        // B = ones => D[m][n] = v_m + v_{m+16} for every n.
        v2f a; a.x = loss; a.y = 0.0f;
        v2f b; b.x = 1.0f; b.y = 1.0f;
        v8f cacc = {};
        v8f d = __builtin_amdgcn_wmma_f32_16x16x4_f32(
            false, a, false, b, (short)0, cacc, false, false);
        // Column-sum the 8 D regs: lanes 0-15 hold rows 0-7, lanes 16-31 rows 8-15.
        s = d[0] + d[1] + d[2] + d[3] + d[4] + d[5] + d[6] + d[7];
        s += __shfl_xor(s, 16, 32);  // combine the two row-halves
    }
#else
    s = loss;
    for (int off = 16; off > 0; off >>= 1) s += __shfl_xor(s, off, 32);
#endif

    const int lane = tid & 31;
    const int wid = tid >> 5;
    if (lane == 0) wave_sums[wid] = s;
    __syncthreads();
    if (tid == 0) {
        float t = 0.0f;
#pragma unroll
        for (int w = 0; w < THREADS / 32; ++w) t += wave_sums[w];
        partial[blockIdx.x] = t;
    }
}

// ---------------------------------------------------------------------------
// Kernel 3: deterministic final reduction of 4092 block partials -> mean.
// ---------------------------------------------------------------------------
__global__ void final_reduce_kernel(const float* __restrict__ partial,
                                    float* __restrict__ out) {
    __shared__ float s[THREADS];
    float t = 0.0f;
    for (int i = threadIdx.x; i < NBLOCKS; i += THREADS) t += partial[i];
    s[threadIdx.x] = t;
    __syncthreads();
    for (int stride = THREADS / 2; stride > 0; stride >>= 1) {
        if (threadIdx.x < stride) s[threadIdx.x] += s[threadIdx.x + stride];
        __syncthreads();
    }
    if (threadIdx.x == 0) out[0] = s[0] * (1.0f / (float)NPAIRS);
}

// ---------------------------------------------------------------------------
extern "C" void kernel_launch(void* const* d_in, const int* in_sizes, int n_in,
                              void* d_out, int out_size, void* d_ws, size_t ws_size,
                              hipStream_t stream) {
    const float* pred  = (const float*)d_in[0];   // (NPAIRS, 27) f32
    const float* alpha = (const float*)d_in[1];   // (27,) f32
    // d_in[2] = obj_gt (unused by the math)
    const int* rel = (const int*)d_in[3];         // (NRELS, 3) i32

    unsigned* bits = (unsigned*)d_ws;                          // NPAIRS u32 ~ 4.19 MB
    float* partial = (float*)((char*)d_ws + (4u << 20));       // 4092 f32 at 4 MB

    (void)hipMemsetAsync(bits, 0, (size_t)NPAIRS * sizeof(unsigned), stream);
    build_mask_kernel<<<(NRELS + THREADS - 1) / THREADS, THREADS, 0, stream>>>(rel, bits);
    row_loss_kernel<<<NBLOCKS, THREADS, 0, stream>>>(pred, alpha, bits, partial);
    final_reduce_kernel<<<1, THREADS, 0, stream>>>(partial, (float*)d_out);
}